// LuminanceLoss_60035052863712
// MI455X (gfx1250) — compile-verified
//
#include <hip/hip_runtime.h>
#include <math.h>

typedef float v2f __attribute__((ext_vector_type(2)));
typedef float v4f __attribute__((ext_vector_type(4)));
typedef float v8f __attribute__((ext_vector_type(8)));

#define HW      (512 * 512)          // 262144 elements per channel plane
#define NPIX    (64 * HW)            // 16,777,216 pixels total
#define NGROUPS (NPIX / 4)           // 4,194,304 float4 groups (HW/4 = 65536 per image)
#define BLOCKS  2048
#define THREADS 256

// ---- CIE-Lab L channel math (fp32, matches skimage constants in reference) ----

__device__ __forceinline__ float srgb_expand(float x) {
    // x in [-1,1] -> s in [0,1] -> inverse sRGB companding
    float s = fmaf(x, 0.5f, 0.5f);
    float a = (s + 0.055f) * (1.0f / 1.055f);       // always > 0.052, log safe
    float p = __powf(a, 2.4f);                       // v_log_f32 + v_exp_f32 path
    return (s > 0.04045f) ? p : s * (1.0f / 12.92f);
}

__device__ __forceinline__ float lab_L(float r, float g, float b) {
    float Y = 0.2126729f * srgb_expand(r)
            + 0.7151522f * srgb_expand(g)
            + 0.0721750f * srgb_expand(b);           // Y in [0,1]
    float croot = __powf(Y, 1.0f / 3.0f);            // Y>eps branch; Y==0 -> 0 (no NaN)
    float lin   = fmaf(7.787f, Y, 16.0f / 116.0f);
    float f     = (Y > 0.008856f) ? croot : lin;
    return fmaf(116.0f, f, -16.0f);
}

// ---- Stage 1: streaming |L_gen - L_tgt| accumulation, WMMA wave reduction ----

__global__ __launch_bounds__(THREADS)
void lum_l1_stage1(const float* __restrict__ gen,
                   const float* __restrict__ tgt,
                   float* __restrict__ block_partials) {
    const int tid    = blockIdx.x * THREADS + threadIdx.x;
    const int stride = BLOCKS * THREADS;

    float acc = 0.0f;
    // NGROUPS % stride == 0 -> every lane runs the same trip count (EXEC all-1s later)
    for (int G = tid; G < NGROUPS; G += stride) {
        const int b = G >> 16;              // HW/4 == 65536 groups per batch image
        const int g = G & 0xFFFF;
        const size_t base = (size_t)b * (3 * HW) + (size_t)g * 4;

        // Non-temporal 128-bit loads: 402 MB stream > 192 MB L2, don't thrash it.
        v4f gr = __builtin_nontemporal_load((const v4f*)(gen + base));
        v4f gg = __builtin_nontemporal_load((const v4f*)(gen + base + HW));
        v4f gb = __builtin_nontemporal_load((const v4f*)(gen + base + 2 * HW));
        v4f tr = __builtin_nontemporal_load((const v4f*)(tgt + base));
        v4f tg = __builtin_nontemporal_load((const v4f*)(tgt + base + HW));
        v4f tb = __builtin_nontemporal_load((const v4f*)(tgt + base + 2 * HW));

        acc += fabsf(lab_L(gr[0], gg[0], gb[0]) - lab_L(tr[0], tg[0], tb[0]));
        acc += fabsf(lab_L(gr[1], gg[1], gb[1]) - lab_L(tr[1], tg[1], tb[1]));
        acc += fabsf(lab_L(gr[2], gg[2], gb[2]) - lab_L(tr[2], tg[2], tb[2]));
        acc += fabsf(lab_L(gr[3], gg[3], gb[3]) - lab_L(tr[3], tg[3], tb[3]));
    }

    // Wave32 reduction via V_WMMA_F32_16X16X4_F32 (full fp32, no precision loss):
    // A (16x4): lane L's VGPR0 -> A[L%16][0 or 2], VGPR1 (=0) -> K=1/3.
    // B = ones(4x16)  =>  D[m][n] = acc(lane m) + acc(lane m+16).
    v2f a;  a[0] = acc;  a[1] = 0.0f;
    v2f bm; bm[0] = 1.0f; bm[1] = 1.0f;
    v8f c = {};
    v8f d = __builtin_amdgcn_wmma_f32_16x16x4_f32(
        /*neg_a=*/false, a, /*neg_b=*/false, bm,
        /*c_mod=*/(short)0, c, /*reuse_a=*/false, /*reuse_b=*/false);

    // Lanes 0-15 see rows M=0..7 in d[0..7]; lanes 16-31 see rows M=8..15.
    float halfsum = d[0] + d[1] + d[2] + d[3] + d[4] + d[5] + d[6] + d[7];
    float wavesum = halfsum + __shfl_xor(halfsum, 16, 32);   // combine both halves

    __shared__ float lds[THREADS / 32];
    const int lane = threadIdx.x & 31;
    const int wave = threadIdx.x >> 5;
    if (lane == 0) lds[wave] = wavesum;
    __syncthreads();
    if (threadIdx.x == 0) {
        float s = 0.0f;
        #pragma unroll
        for (int i = 0; i < THREADS / 32; ++i) s += lds[i];
        block_partials[blockIdx.x] = s;      // deterministic: no atomics
    }
}

// ---- Stage 2: deterministic final reduction in fp64, writes the mean ----

__global__ __launch_bounds__(256)
void lum_l1_stage2(const float* __restrict__ block_partials,
                   float* __restrict__ out) {
    __shared__ double lds[256];
    double s = 0.0;
    for (int i = threadIdx.x; i < BLOCKS; i += 256) s += (double)block_partials[i];
    lds[threadIdx.x] = s;
    __syncthreads();
    for (int off = 128; off > 0; off >>= 1) {
        if (threadIdx.x < off) lds[threadIdx.x] += lds[threadIdx.x + off];
        __syncthreads();
    }
    if (threadIdx.x == 0) {
        out[0] = (float)(lds[0] / (double)NPIX);   // WEIGHT == 1.0
    }
}

extern "C" void kernel_launch(void* const* d_in, const int* in_sizes, int n_in,
                              void* d_out, int out_size, void* d_ws, size_t ws_size,
                              hipStream_t stream) {
    const float* gen = (const float*)d_in[0];
    const float* tgt = (const float*)d_in[1];
    float* out       = (float*)d_out;
    float* partials  = (float*)d_ws;        // BLOCKS floats = 8 KB scratch

    lum_l1_stage1<<<BLOCKS, THREADS, 0, stream>>>(gen, tgt, partials);
    lum_l1_stage2<<<1, 256, 0, stream>>>(partials, out);
}